// SSE_41308995452948
// MI455X (gfx1250) — compile-verified
//
#include <hip/hip_runtime.h>
#include <math.h>

// ---------------------------------------------------------------------------
// SSE fixed-point GNN for MI455X (gfx1250), wave32 + V_WMMA_F32_16X16X4_F32
// ---------------------------------------------------------------------------
typedef __attribute__((ext_vector_type(2))) float v2f;
typedef __attribute__((ext_vector_type(8))) float v8f;

#define TPB 256
static constexpr float ALPHA_C = 0.1f;
static constexpr float TOL2_C  = 1e-10f;   // (1e-5)^2
static constexpr int   MAX_IT  = 20;

// -------------------------------------------------- init / zero helpers
__global__ void k_init_flag(int* flag) {
    if (blockIdx.x == 0 && threadIdx.x == 0) *flag = 0;
}

__global__ void k_zero(float* __restrict__ p, int n) {
    int i = blockIdx.x * blockDim.x + threadIdx.x;
    if (i < n) p[i] = 0.f;
}

// zero agg (guarded by convergence flag) and reset gsum for this iteration
__global__ void k_zero_guard(float* __restrict__ p, int n,
                             const int* __restrict__ flag,
                             float* __restrict__ gsum) {
    if (*flag) return;
    int i = blockIdx.x * blockDim.x + threadIdx.x;
    if (i == 0) *gsum = 0.f;
    if (i < n) p[i] = 0.f;
}

// -------------------------------------------------- static edge aggregation
// aggS[dst][0:16] += nf[src];  aggS[dst][16:24] += ef[e]
__global__ void k_scatter_static(const float* __restrict__ nf,
                                 const float* __restrict__ ef,
                                 const int*   __restrict__ src,
                                 const int*   __restrict__ dst,
                                 float*       __restrict__ aggS, int E) {
    int idx = blockIdx.x * blockDim.x + threadIdx.x;
    int e = idx >> 3, j = idx & 7;
    if (e >= E || j >= 6) return;
    int s = src[e], d = dst[e];
    if (j < 4) {
        const float4 v = *(const float4*)(nf + (long)s * 16 + j * 4);
        float* o = aggS + (long)d * 24 + j * 4;
        atomicAdd(o + 0, v.x); atomicAdd(o + 1, v.y);
        atomicAdd(o + 2, v.z); atomicAdd(o + 3, v.w);
    } else {
        int c = (j - 4) * 4;
        const float4 v = *(const float4*)(ef + (long)e * 8 + c);
        float* o = aggS + (long)d * 24 + 16 + c;
        atomicAdd(o + 0, v.x); atomicAdd(o + 1, v.y);
        atomicAdd(o + 2, v.z); atomicAdd(o + 3, v.w);
    }
}

// Bstatic[i] = b1 + nf[i] @ W1[0:16] + aggS[i] @ W1[48:72]
__global__ void k_bstatic(const float* __restrict__ nf,
                          const float* __restrict__ aggS,
                          const float* __restrict__ W1,
                          const float* __restrict__ b1,
                          float*       __restrict__ Bst, int Nn) {
    int i = blockIdx.x * blockDim.x + threadIdx.x;
    if (i >= Nn) return;
    float acc[32];
#pragma unroll
    for (int n = 0; n < 32; ++n) acc[n] = b1[n];
    for (int k = 0; k < 16; ++k) {
        float x = nf[(long)i * 16 + k];
        const float* wr = W1 + k * 32;
#pragma unroll
        for (int n = 0; n < 32; ++n) acc[n] = fmaf(x, wr[n], acc[n]);
    }
    for (int k = 0; k < 24; ++k) {
        float x = aggS[(long)i * 24 + k];
        const float* wr = W1 + (48 + k) * 32;
#pragma unroll
        for (int n = 0; n < 32; ++n) acc[n] = fmaf(x, wr[n], acc[n]);
    }
#pragma unroll
    for (int n = 0; n < 32; ++n) Bst[(long)i * 32 + n] = acc[n];
}

// -------------------------------------------------- per-iteration scatter
// agg[dst] += h[src]   (32 floats/edge; 8 threads/edge, float4 each)
__global__ void k_scatter_h(const float* __restrict__ h,
                            const int*   __restrict__ src,
                            const int*   __restrict__ dst,
                            float*       __restrict__ agg, int E,
                            const int*   __restrict__ flag, int guard) {
    if (guard && *flag) return;
    int idx = blockIdx.x * blockDim.x + threadIdx.x;
    int e = idx >> 3;
    if (e >= E) return;
    int c = (idx & 7) * 4;
    int s = src[e], d = dst[e];
    const float4 v = *(const float4*)(h + (long)s * 32 + c);
    float* o = agg + (long)d * 32 + c;
    atomicAdd(o + 0, v.x); atomicAdd(o + 1, v.y);
    atomicAdd(o + 2, v.z); atomicAdd(o + 3, v.w);
}

// -------------------------------------------------- WMMA GEMM + epilogue
// t = relu(agg @ W1h + Bstatic)
// mode 0: hout = (1-a)*h + a*t, accumulate ||h-hout||^2 into gsum
// mode 1: hout = t            (final sse_layer)
__global__ void k_gemm(const float* __restrict__ agg,
                       const float* __restrict__ W1h,
                       const float* __restrict__ Bst,
                       const float* __restrict__ hcur,
                       float*       __restrict__ hout,
                       float*       __restrict__ gsum,
                       int Nn, int mode, const int* __restrict__ flag) {
    if (mode == 0 && *flag) return;
    const int lane  = threadIdx.x & 31;
    const int wave  = threadIdx.x >> 5;
    const int m0    = (blockIdx.x * (blockDim.x >> 5) + wave) * 16;
    if (m0 >= Nn) return;

    const int khalf = (lane >> 4) << 1;   // 0 or 2 (K offset within group of 4)
    const int nlo   = lane & 15;          // column within 16-wide tile

    // A fragments: clamp the row instead of predicating the load. Rows >= Nn
    // feed only output rows >= Nn, which the epilogue never stores, so the
    // clamped (garbage) data is harmless and we avoid exec-mask branching.
    const int am  = m0 + nlo;
    const int amc = (am < Nn) ? am : (Nn - 1);
    const float* arow = agg + (long)amc * 32;
    v2f a[8];
#pragma unroll
    for (int kk = 0; kk < 8; ++kk) {
        const float2 t = *(const float2*)(arow + kk * 4 + khalf);
        a[kk].x = t.x; a[kk].y = t.y;
    }

    // B fragments: W1h is [32 x 32] row-major; B[k][n], n split into 2 tiles
    v2f Bf[8][2];
#pragma unroll
    for (int kk = 0; kk < 8; ++kk) {
#pragma unroll
        for (int nt = 0; nt < 2; ++nt) {
            const int kr  = kk * 4 + khalf;
            const int col = nt * 16 + nlo;
            v2f b;
            b.x = W1h[(kr + 0) * 32 + col];
            b.y = W1h[(kr + 1) * 32 + col];
            Bf[kk][nt] = b;
        }
    }

    v8f c0 = {}; v8f c1 = {};
#pragma unroll
    for (int kk = 0; kk < 8; ++kk) {
        c0 = __builtin_amdgcn_wmma_f32_16x16x4_f32(false, a[kk], false, Bf[kk][0],
                                                   (short)0, c0, false, false);
        c1 = __builtin_amdgcn_wmma_f32_16x16x4_f32(false, a[kk], false, Bf[kk][1],
                                                   (short)0, c1, false, false);
    }

    // epilogue: C layout -> VGPR r holds rows m0+r (lanes 0-15) / m0+8+r (16-31)
    float lsum = 0.f;
    const int rbase = (lane >> 4) << 3;   // 0 or 8
#pragma unroll
    for (int r = 0; r < 8; ++r) {
        const int row = m0 + r + rbase;
        if (row >= Nn) continue;
        const long base = (long)row * 32;
        // tile 0 : cols 0..15
        {
            float t = c0[r] + Bst[base + nlo];
            t = fmaxf(t, 0.f);
            if (mode == 0) {
                float hp = hcur[base + nlo];
                float hn = (1.f - ALPHA_C) * hp + ALPHA_C * t;
                float d  = hp - hn;
                lsum = fmaf(d, d, lsum);
                hout[base + nlo] = hn;
            } else hout[base + nlo] = t;
        }
        // tile 1 : cols 16..31
        {
            const int col = 16 + nlo;
            float t = c1[r] + Bst[base + col];
            t = fmaxf(t, 0.f);
            if (mode == 0) {
                float hp = hcur[base + col];
                float hn = (1.f - ALPHA_C) * hp + ALPHA_C * t;
                float d  = hp - hn;
                lsum = fmaf(d, d, lsum);
                hout[base + col] = hn;
            } else hout[base + col] = t;
        }
    }

    if (mode == 0) {
#pragma unroll
        for (int off = 16; off > 0; off >>= 1)
            lsum += __shfl_xor(lsum, off, 32);
        if (lane == 0) atomicAdd(gsum, lsum);
    }
}

// -------------------------------------------------- commit / convergence
__global__ void k_commit(float* __restrict__ h, const float* __restrict__ hn,
                         const float* __restrict__ gsum, int* __restrict__ flag,
                         int total) {
    if (*flag) return;
    const float g2 = *gsum;
    int i = blockIdx.x * blockDim.x + threadIdx.x;
    if (g2 < TOL2_C) {               // converged: keep h, stop all later work
        if (i == 0) *flag = 1;
        return;
    }
    if (i < total) h[i] = hn[i];
}

// -------------------------------------------------- regressor MLP
// out[i] = relu(hf[i] @ Wr1 + br1) @ Wr2 + br2
__global__ void k_regress(const float* __restrict__ hf,
                          const float* __restrict__ Wr1,
                          const float* __restrict__ br1,
                          const float* __restrict__ Wr2,
                          const float* __restrict__ br2,
                          float*       __restrict__ out, int Nn) {
    int i = blockIdx.x * blockDim.x + threadIdx.x;
    if (i >= Nn) return;
    float hv[32];
#pragma unroll
    for (int k = 0; k < 32; ++k) hv[k] = hf[(long)i * 32 + k];
    float o = br2[0];
    for (int n = 0; n < 32; ++n) {
        float u = br1[n];
#pragma unroll
        for (int k = 0; k < 32; ++k) u = fmaf(hv[k], Wr1[k * 32 + n], u);
        u = fmaxf(u, 0.f);
        o = fmaf(u, Wr2[n], o);
    }
    out[i] = o;
}

// ---------------------------------------------------------------------------
extern "C" void kernel_launch(void* const* d_in, const int* in_sizes, int n_in,
                              void* d_out, int out_size, void* d_ws, size_t ws_size,
                              hipStream_t stream) {
    const float* nf  = (const float*)d_in[0];
    const float* ef  = (const float*)d_in[1];
    const int*   src = (const int*)  d_in[2];
    const int*   dst = (const int*)  d_in[3];
    const float* W1  = (const float*)d_in[4];
    const float* b1  = (const float*)d_in[5];
    const float* Wr1 = (const float*)d_in[6];
    const float* br1 = (const float*)d_in[7];
    const float* Wr2 = (const float*)d_in[8];
    const float* br2 = (const float*)d_in[9];

    const int Nn = in_sizes[0] / 16;   // nodes
    const int Ee = in_sizes[2];        // edges
    const float* W1h = W1 + 16 * 32;   // rows 16..47 of W1 (the h-dependent block)

    // workspace layout (floats)
    float* ws    = (float*)d_ws;
    float* h     = ws;
    float* hnext = h     + (size_t)Nn * 32;
    float* agg   = hnext + (size_t)Nn * 32;
    float* Bst   = agg   + (size_t)Nn * 32;
    float* aggS  = Bst   + (size_t)Nn * 32;
    float* gsum  = aggS  + (size_t)Nn * 24;
    int*   flag  = (int*)(gsum + 1);

    const int gN32  = (Nn * 32 + TPB - 1) / TPB;
    const int gN24  = (Nn * 24 + TPB - 1) / TPB;
    const int gNode = (Nn + TPB - 1) / TPB;
    const int gEdge = (Ee * 8 + TPB - 1) / TPB;
    const int waves = (Nn + 15) / 16;
    const int gGemm = (waves + 7) / 8;          // 8 waves (256 threads) / block

    // ---- init + static precompute (every call: fully deterministic)
    k_init_flag<<<1, 1, 0, stream>>>(flag);
    k_zero<<<gN32, TPB, 0, stream>>>(h,    Nn * 32);
    k_zero<<<gN24, TPB, 0, stream>>>(aggS, Nn * 24);
    k_scatter_static<<<gEdge, TPB, 0, stream>>>(nf, ef, src, dst, aggS, Ee);
    k_bstatic<<<gNode, TPB, 0, stream>>>(nf, aggS, W1, b1, Bst, Nn);

    // ---- fixed-point iterations (flag-guarded; at most MAX_IT commits)
    for (int it = 0; it < MAX_IT; ++it) {
        k_zero_guard<<<gN32, TPB, 0, stream>>>(agg, Nn * 32, flag, gsum);
        k_scatter_h<<<gEdge, TPB, 0, stream>>>(h, src, dst, agg, Ee, flag, 1);
        k_gemm<<<gGemm, TPB, 0, stream>>>(agg, W1h, Bst, h, hnext, gsum,
                                          Nn, /*mode=*/0, flag);
        k_commit<<<gN32, TPB, 0, stream>>>(h, hnext, gsum, flag, Nn * 32);
    }

    // ---- final differentiable sse_layer (always runs)
    k_zero<<<gN32, TPB, 0, stream>>>(agg, Nn * 32);
    k_scatter_h<<<gEdge, TPB, 0, stream>>>(h, src, dst, agg, Ee, flag, 0);
    k_gemm<<<gGemm, TPB, 0, stream>>>(agg, W1h, Bst, h, hnext, gsum,
                                      Nn, /*mode=*/1, flag);

    // ---- regressor MLP -> d_out [N,1]
    k_regress<<<gNode, TPB, 0, stream>>>(hnext, Wr1, br1, Wr2, br2,
                                         (float*)d_out, Nn);
}